// PermutationCrossEntropy_82076825026789
// MI455X (gfx1250) — compile-verified
//
#include <hip/hip_runtime.h>
#include <hip/hip_bf16.h>

typedef __attribute__((ext_vector_type(2)))  float v2f;
typedef __attribute__((ext_vector_type(4)))  float v4f;
typedef __attribute__((ext_vector_type(8)))  float v8f;

#define NSPK 4
#define NPERM 24

// ---------------------------------------------------------------------------
// Branchless permutation indicator mask.
// Lexicographic permutation p of {0,1,2,3} via factorial number system.
// Returns 16-bit mask with bit (4*q + PERM[p][q]) set; 0 for p >= 24.
// ---------------------------------------------------------------------------
__device__ __forceinline__ int nthSetBit4(unsigned m, int k) {
    int res = 0, cnt = 0;
    #pragma unroll
    for (int bit = 0; bit < 4; ++bit) {
        int set = (int)((m >> bit) & 1u);
        res = (set && (cnt == k)) ? bit : res;
        cnt += set;
    }
    return res;
}

__device__ __forceinline__ unsigned permMask16(int p) {
    int a = p / 6;
    int r = p - 6 * a;
    int bq = r >> 1;
    int cq = r & 1;
    unsigned m = 0xFu;
    int d0 = nthSetBit4(m, a);  m &= ~(1u << d0);
    int d1 = nthSetBit4(m, bq); m &= ~(1u << d1);
    int d2 = nthSetBit4(m, cq); m &= ~(1u << d2);
    int d3 = __builtin_ctz(m);              // exactly one bit left
    unsigned mask = (1u << d0) | (1u << (4 + d1)) | (1u << (8 + d2)) | (1u << (12 + d3));
    return (p < NPERM) ? mask : 0u;
}

// ---------------------------------------------------------------------------
// Kernel 1: one wave per (b, s) row. Streaming NT b128 loads (2 per iter),
// online log-sum-exp (one rescale per 8 elems), butterfly wave reduction.
// logZ[row] = max + log(sum exp).
// ---------------------------------------------------------------------------
__global__ __launch_bounds__(256) void pit_logz(const float* __restrict__ preds,
                                                float* __restrict__ logZ,
                                                int C, int nRows) {
    int row  = (int)((blockIdx.x * blockDim.x + threadIdx.x) >> 5);
    int lane = (int)(threadIdx.x & 31);
    if (row >= nRows) return;

    const float* rp  = preds + (size_t)row * (size_t)C;
    const v4f*   rv  = (const v4f*)rp;
    int          nv4 = C >> 2;

    float m = -__builtin_inff();
    float s = 0.0f;

    int idx = lane;
    for (; idx + 32 < nv4; idx += 64) {
        v4f v0 = __builtin_nontemporal_load(rv + idx);        // global_load_b128 NT
        v4f v1 = __builtin_nontemporal_load(rv + idx + 32);
        float cm = fmaxf(fmaxf(fmaxf(v0.x, v0.y), fmaxf(v0.z, v0.w)),
                         fmaxf(fmaxf(v1.x, v1.y), fmaxf(v1.z, v1.w)));
        float nm = fmaxf(m, cm);
        s = s * __expf(m - nm)
            + __expf(v0.x - nm) + __expf(v0.y - nm) + __expf(v0.z - nm) + __expf(v0.w - nm)
            + __expf(v1.x - nm) + __expf(v1.y - nm) + __expf(v1.z - nm) + __expf(v1.w - nm);
        m = nm;
    }
    if (idx < nv4) {
        v4f v = __builtin_nontemporal_load(rv + idx);
        float cm = fmaxf(fmaxf(v.x, v.y), fmaxf(v.z, v.w));
        float nm = fmaxf(m, cm);
        s = s * __expf(m - nm) + __expf(v.x - nm) + __expf(v.y - nm)
            + __expf(v.z - nm) + __expf(v.w - nm);
        m = nm;
    }
    int base = nv4 << 2;
    int rem  = C - base;
    if (lane < rem) {
        float v  = rp[base + lane];
        float nm = fmaxf(m, v);
        s = s * __expf(m - nm) + __expf(v - nm);
        m = nm;
    }

    // wave32 butterfly combine of (m, s)
    #pragma unroll
    for (int off = 16; off >= 1; off >>= 1) {
        float om = __shfl_xor(m, off, 32);
        float os = __shfl_xor(s, off, 32);
        float nm = fmaxf(m, om);
        s = s * __expf(m - nm) + os * __expf(om - nm);
        m = nm;
    }
    if (lane == 0) logZ[row] = m + __logf(s);
}

// ---------------------------------------------------------------------------
// Kernel 2: one wave per 16 batch elements.
//   costflat[b][4i+j] = logZ[b,i] - preds[b, i, t_j]
//   per_perm[b,p]     = costflat x P^T  via V_WMMA_F32_16X16X4_F32
// (4 accumulating K=4 steps; perms 0-15 in d0, perms 16-23 in d1).
// Min over perms, sum 16 losses, one partial per wave.
// ---------------------------------------------------------------------------
__global__ __launch_bounds__(256) void pit_cost_wmma(
        const float* __restrict__ preds, const int* __restrict__ targets,
        const float* __restrict__ logZ, float* __restrict__ tileSums,
        int C, int nTiles) {
    int wv   = (int)((blockIdx.x * blockDim.x + threadIdx.x) >> 5);
    int lane = (int)(threadIdx.x & 31);
    if (wv >= nTiles) return;           // wave-uniform; active waves keep EXEC all-1s

    const int h = lane >> 4;            // wave half
    const int n = lane & 15;            // M row (A) / N col (B,D)
    const int b = wv * 16 + n;          // batch element owned by this lane's A rows

    // Branchless per-lane permutation indicator masks (no table, no loads).
    const unsigned mask0 = permMask16(n);        // perms 0..15
    const unsigned mask1 = permMask16(16 + n);   // perms 16..23; 0 for n >= 8

    int   tj[NSPK];
    float lz[NSPK];
    #pragma unroll
    for (int j = 0; j < NSPK; ++j) tj[j] = targets[b * NSPK + j];
    #pragma unroll
    for (int i = 0; i < NSPK; ++i) lz[i] = logZ[b * NSPK + i];

    // Issue all 8 gathers first, then compute cost fragments.
    // Lane covers kk = 4q + 2h + e  (q = speaker i, jc = 2h+e = target column j).
    float pv[8];
    #pragma unroll
    for (int q = 0; q < NSPK; ++q) {
        #pragma unroll
        for (int e = 0; e < 2; ++e) {
            int jc = 2 * h + e;
            pv[2 * q + e] = preds[((size_t)b * NSPK + q) * (size_t)C + tj[jc]];
        }
    }
    float cf[8];
    #pragma unroll
    for (int q = 0; q < NSPK; ++q) {
        cf[2 * q]     = lz[q] - pv[2 * q];
        cf[2 * q + 1] = lz[q] - pv[2 * q + 1];
    }

    v8f d0 = {}; // per_perm for perms 0..15
    v8f d1 = {}; // per_perm for perms 16..23 (cols 8..15 zero-padded)

    // A 16x4 f32 layout: lane (M=n) VGPR0 = K=2h, VGPR1 = K=2h+1; B mirrored.
    #pragma unroll
    for (int q = 0; q < 4; ++q) {
        int kkA = 4 * q + 2 * h;
        v2f a, b0, b1;
        a.x  = cf[2 * q];
        a.y  = cf[2 * q + 1];
        b0.x = ((mask0 >> kkA) & 1u)       ? 1.0f : 0.0f;
        b0.y = ((mask0 >> (kkA + 1)) & 1u) ? 1.0f : 0.0f;
        b1.x = ((mask1 >> kkA) & 1u)       ? 1.0f : 0.0f;
        b1.y = ((mask1 >> (kkA + 1)) & 1u) ? 1.0f : 0.0f;
        d0 = __builtin_amdgcn_wmma_f32_16x16x4_f32(false, a, false, b0,
                                                   (short)0, d0, false, false);
        d1 = __builtin_amdgcn_wmma_f32_16x16x4_f32(false, a, false, b1,
                                                   (short)0, d1, false, false);
    }

    // D layout: VGPR r, lane<16 -> (M=r, N=lane); lane>=16 -> (M=r+8, N=lane-16).
    // Min over perm columns within each 16-lane half; accumulate 8 losses/half.
    const float INF = __builtin_inff();
    float sumh = 0.0f;
    #pragma unroll
    for (int r = 0; r < 8; ++r) {
        float v0 = d0[r];
        float v1 = (n < 8) ? d1[r] : INF;   // cols 8..15 of d1 are padding (==0)
        #pragma unroll
        for (int mask = 8; mask >= 1; mask >>= 1) {
            v0 = fminf(v0, __shfl_xor(v0, mask, 32));
            v1 = fminf(v1, __shfl_xor(v1, mask, 32));
        }
        sumh += fminf(v0, v1);              // loss of batch element m = r + 8*h
    }
    float tot = sumh + __shfl_xor(sumh, 16, 32);
    if (lane == 0) tileSums[wv] = tot;
}

// ---------------------------------------------------------------------------
// Kernel 3: deterministic fixed-order tree reduction -> mean.
// ---------------------------------------------------------------------------
__global__ __launch_bounds__(256) void pit_finalize(const float* __restrict__ tileSums,
                                                    float* __restrict__ out,
                                                    int nTiles, float invB) {
    __shared__ float sm[256];
    int tid = (int)threadIdx.x;
    float v = 0.0f;
    for (int i = tid; i < nTiles; i += 256) v += tileSums[i];
    sm[tid] = v;
    __syncthreads();
    #pragma unroll
    for (int s2 = 128; s2 > 0; s2 >>= 1) {
        if (tid < s2) sm[tid] += sm[tid + s2];
        __syncthreads();
    }
    if (tid == 0) out[0] = sm[0] * invB;
}

extern "C" void kernel_launch(void* const* d_in, const int* in_sizes, int n_in,
                              void* d_out, int out_size, void* d_ws, size_t ws_size,
                              hipStream_t stream) {
    const float* preds   = (const float*)d_in[0];
    const int*   targets = (const int*)d_in[1];   // harness passes integer inputs as int

    const int nRows  = in_sizes[1];               // B * S = 16384
    const int B      = nRows / NSPK;              // 4096
    const int C      = in_sizes[0] / nRows;       // 5000
    const int nTiles = B / 16;                    // 256

    float* logZ     = (float*)d_ws;               // nRows floats
    float* tileSums = logZ + nRows;               // nTiles floats

    // Kernel 1: 1 wave per row (bandwidth-bound, ~327 MB @ 23.3 TB/s)
    int blkB = (nRows * 32 + 255) / 256;
    pit_logz<<<blkB, 256, 0, stream>>>(preds, logZ, C, nRows);

    // Kernel 2: 1 wave per 16 batch elements (WMMA per-perm sums)
    int blkC = (nTiles * 32 + 255) / 256;
    pit_cost_wmma<<<blkC, 256, 0, stream>>>(preds, targets, logZ, tileSums, C, nTiles);

    // Kernel 3: mean
    pit_finalize<<<1, 256, 0, stream>>>(tileSums, (float*)d_out, nTiles, 1.0f / (float)B);
}